// LambdaLayer_79199196938288
// MI455X (gfx1250) — compile-verified
//
#include <hip/hip_runtime.h>

typedef __attribute__((ext_vector_type(2))) float v2f;
typedef __attribute__((ext_vector_type(8))) float v8f;

#define N_POS 4096
#define DDIM  256
#define EPS_F 1e-5f

// ---------------------------------------------------------------------------
// Kernel 1: fused q/k/v projection GEMM + BN.
// Stacked weights: ch 0..63 = Wq rows, 64..79 = Wk rows, 80..143 = Wv rows.
// Block: 256 threads (8 waves). Each wave owns a 16-wide n strip, computes all
// 9 row-tiles (144 channels) with fp32 WMMA 16x16x4, K=256 (64 steps).
// ---------------------------------------------------------------------------
__global__ __launch_bounds__(256) void proj_kernel(
    const float* __restrict__ x,
    const float* __restrict__ Wq, const float* __restrict__ Wk, const float* __restrict__ Wv,
    const float* __restrict__ qg, const float* __restrict__ qb,
    const float* __restrict__ qm, const float* __restrict__ qv,
    const float* __restrict__ vg, const float* __restrict__ vb,
    const float* __restrict__ vm, const float* __restrict__ vvr,
    float* __restrict__ qbuf, float* __restrict__ kbuf, float* __restrict__ vvbuf)
{
  __shared__ float sW[144 * 260];   // row stride 260: bank-friendly + 8B aligned pairs
  __shared__ float sScale[144];
  __shared__ float sBias[144];

  const int b    = blockIdx.y;
  const int tid  = threadIdx.x;
  const int wave = tid >> 5;
  const int lane = tid & 31;
  const int n0   = blockIdx.x * 128 + wave * 16;

  // Stage stacked weights in LDS (149 KB, fits CDNA5 320KB WGP LDS)
  for (int i = tid; i < 144 * 256; i += 256) {
    int ch = i >> 8, k = i & 255;
    float w;
    if (ch < 64)      w = Wq[ch * 256 + k];
    else if (ch < 80) w = Wk[(ch - 64) * 256 + k];
    else              w = Wv[(ch - 80) * 256 + k];
    sW[ch * 260 + k] = w;
  }
  if (tid < 144) {
    float s, bb;
    if (tid < 64)      { s = qg[tid] * rsqrtf(qv[tid] + EPS_F); bb = qb[tid] - qm[tid] * s; }
    else if (tid < 80) { s = 1.f; bb = 0.f; }
    else               { int c = tid - 80; s = vg[c] * rsqrtf(vvr[c] + EPS_F); bb = vb[c] - vm[c] * s; }
    sScale[tid] = s; sBias[tid] = bb;
  }
  __syncthreads();

  const float* xb  = x + (size_t)b * DDIM * N_POS;
  const int cN  = lane & 15;          // A-row / B-col within tile
  const int khi = (lane >> 4) << 1;   // upper half-wave handles K+2,K+3

  v8f acc[9];
  #pragma unroll
  for (int m = 0; m < 9; ++m) acc[m] = (v8f){0.f,0.f,0.f,0.f,0.f,0.f,0.f,0.f};

  for (int s = 0; s < 64; ++s) {
    const int kb = 4 * s + khi;
    v2f bf;   // B: K = kb+j, N = n0+cN  (x read exactly once from HBM)
    bf.x = xb[(size_t)(kb + 0) * N_POS + n0 + cN];
    bf.y = xb[(size_t)(kb + 1) * N_POS + n0 + cN];
    #pragma unroll
    for (int m = 0; m < 9; ++m) {
      v2f af;   // A: M = cN (channel within tile), K = kb+j
      af.x = sW[(m * 16 + cN) * 260 + kb + 0];
      af.y = sW[(m * 16 + cN) * 260 + kb + 1];
      acc[m] = __builtin_amdgcn_wmma_f32_16x16x4_f32(
          false, af, false, bf, (short)0, acc[m], false, false);
    }
  }

  // Epilogue: BN + scatter to channel-major buffers (coalesced along n)
  const int hi8 = (lane >> 4) << 3;
  #pragma unroll
  for (int m = 0; m < 9; ++m) {
    #pragma unroll
    for (int r = 0; r < 8; ++r) {
      int ch = m * 16 + r + hi8;
      int n  = n0 + cN;
      float val = acc[m][r] * sScale[ch] + sBias[ch];
      if (ch < 64)       qbuf [((size_t)b * 64 + ch)        * N_POS + n] = val;
      else if (ch < 80)  kbuf [((size_t)b * 16 + (ch - 64)) * N_POS + n] = val;
      else               vvbuf[((size_t)b * 64 + (ch - 80)) * N_POS + n] = val;
    }
  }
}

// ---------------------------------------------------------------------------
// Kernel 2: softmax over positions (one block per (b, k-channel) row of 4096)
// ---------------------------------------------------------------------------
__global__ __launch_bounds__(256) void softmax_kernel(float* __restrict__ kbuf)
{
  float* p = kbuf + (size_t)blockIdx.x * N_POS;
  __shared__ float red[256];
  const int tid = threadIdx.x;

  float mx = -3.4e38f;
  for (int i = tid; i < N_POS; i += 256) mx = fmaxf(mx, p[i]);
  red[tid] = mx; __syncthreads();
  for (int s = 128; s > 0; s >>= 1) { if (tid < s) red[tid] = fmaxf(red[tid], red[tid + s]); __syncthreads(); }
  mx = red[0]; __syncthreads();

  float sum = 0.f;
  for (int i = tid; i < N_POS; i += 256) { float e = __expf(p[i] - mx); p[i] = e; sum += e; }
  red[tid] = sum; __syncthreads();
  for (int s = 128; s > 0; s >>= 1) { if (tid < s) red[tid] += red[tid + s]; __syncthreads(); }
  float inv = 1.f / red[0];
  for (int i = tid; i < N_POS; i += 256) p[i] *= inv;
}

// ---------------------------------------------------------------------------
// Kernel 3: lam_c'[b,k,v] = conv_b[k] + sum_n kk[b,k,n] * vv[b,v,n]
// LDS-tiled over n; one block per batch; each thread owns 4 (k,v) pairs.
// ---------------------------------------------------------------------------
__global__ __launch_bounds__(256) void lamc_kernel(
    const float* __restrict__ kk, const float* __restrict__ vv,
    const float* __restrict__ conv_b, float* __restrict__ lamc)
{
  __shared__ float sk[16 * 129];
  __shared__ float sv[64 * 129];
  const int b = blockIdx.x, tid = threadIdx.x;
  const float* kp = kk + (size_t)b * 16 * N_POS;
  const float* vp = vv + (size_t)b * 64 * N_POS;

  float acc4[4] = {0.f, 0.f, 0.f, 0.f};
  for (int c = 0; c < N_POS; c += 128) {
    __syncthreads();
    for (int i = tid; i < 16 * 128; i += 256) { int r = i >> 7, t = i & 127; sk[r * 129 + t] = kp[(size_t)r * N_POS + c + t]; }
    for (int i = tid; i < 64 * 128; i += 256) { int r = i >> 7, t = i & 127; sv[r * 129 + t] = vp[(size_t)r * N_POS + c + t]; }
    __syncthreads();
    #pragma unroll
    for (int j = 0; j < 4; ++j) {
      int idx = tid + j * 256, k = idx >> 6, v = idx & 63;
      float a = 0.f;
      for (int t = 0; t < 128; ++t) a += sk[k * 129 + t] * sv[v * 129 + t];
      acc4[j] += a;
    }
  }
  #pragma unroll
  for (int j = 0; j < 4; ++j) {
    int idx = tid + j * 256, k = idx >> 6;
    lamc[(size_t)b * 1024 + idx] = acc4[j] + conv_b[k];
  }
}

// ---------------------------------------------------------------------------
// Kernel 4: out[b,n,h,v] = sum_k q*(lam_c' + conv-term), all via WMMA.
//   Term1: Q_tile(16pos x 16k) x lam_c'(16k x 64v)
//   Term2: banded GEMM Gt(16 x 40) x VVwin(38 x 64) with g = Q x Wconv^T.
// Block: one batch, 64-position tile, 8 waves x 2 tasks (4 heads x 4 subtiles).
// ---------------------------------------------------------------------------
__global__ __launch_bounds__(256) void out_kernel(
    const float* __restrict__ qbuf, const float* __restrict__ vvbuf,
    const float* __restrict__ lamc, const float* __restrict__ conv_w,
    float* __restrict__ out)
{
  __shared__ float sQ  [64 * 80];    // [ch][pos]
  __shared__ float sVV [86 * 80];    // [window row j][v], j = n0-11+j
  __shared__ float sLam[16 * 80];    // [k][v]
  __shared__ float sCW [16 * 24];    // [k][r]
  __shared__ float sG  [8 * 576];    // per-wave g, layout [r][pos], stride 24
  __shared__ float sOut[256 * 68];   // [channel][pos]

  const int b   = blockIdx.y;
  const int n0  = blockIdx.x * 64;
  const int tid = threadIdx.x;
  const int wave = tid >> 5, lane = tid & 31;
  const int cN  = lane & 15;
  const int khi = (lane >> 4) << 1;
  const int hi8 = (lane >> 4) << 3;

  // Stage Q tile (64 ch x 64 pos), transposed VV window, lam_c', conv_w
  for (int i = tid; i < 64 * 64; i += 256) {
    int ch = i >> 6, pos = i & 63;
    sQ[ch * 80 + pos] = qbuf[((size_t)b * 64 + ch) * N_POS + n0 + pos];
  }
  for (int i = tid; i < 64 * 86; i += 256) {
    int v = i / 86, j = i % 86;
    int n = n0 - 11 + j;
    sVV[j * 80 + v] = (n >= 0 && n < N_POS) ? vvbuf[((size_t)b * 64 + v) * N_POS + n] : 0.f;
  }
  for (int i = tid; i < 16 * 64; i += 256) {
    int k = i >> 6, v = i & 63;
    sLam[k * 80 + v] = lamc[(size_t)b * 1024 + k * 64 + v];
  }
  for (int i = tid; i < 16 * 23; i += 256) sCW[(i / 23) * 24 + (i % 23)] = conv_w[i];
  __syncthreads();

  for (int task = wave; task < 16; task += 8) {
    const int h  = task >> 2;
    const int ps = (task & 3) * 16;
    const int gbase = wave * 576;

    // g[pos][r] = sum_k q[h,pos,k] * conv_w[k][r]  (stored [r][pos], stride 24)
    for (int idx = lane; idx < 23 * 16; idx += 32) {
      int r = idx >> 4, pp = idx & 15;
      float s = 0.f;
      #pragma unroll
      for (int k = 0; k < 16; ++k) s += sQ[(h * 16 + k) * 80 + ps + pp] * sCW[k * 24 + r];
      sG[gbase + r * 24 + pp] = s;   // LDS ops in-order within wave
    }

    v8f acc[4];
    #pragma unroll
    for (int vt = 0; vt < 4; ++vt) acc[vt] = (v8f){0.f,0.f,0.f,0.f,0.f,0.f,0.f,0.f};

    // GEMM1: Q(ps..ps+15, k=0..15) x lam_c'(k, v)
    #pragma unroll
    for (int s = 0; s < 4; ++s) {
      const int kb = 4 * s + khi;
      v2f af;
      af.x = sQ[(h * 16 + kb + 0) * 80 + ps + cN];
      af.y = sQ[(h * 16 + kb + 1) * 80 + ps + cN];
      #pragma unroll
      for (int vt = 0; vt < 4; ++vt) {
        v2f bf;
        bf.x = sLam[(kb + 0) * 80 + vt * 16 + cN];
        bf.y = sLam[(kb + 1) * 80 + vt * 16 + cN];
        acc[vt] = __builtin_amdgcn_wmma_f32_16x16x4_f32(
            false, af, false, bf, (short)0, acc[vt], false, false);
      }
    }

    // GEMM2: banded conv term. out2[i,v] = sum_j Gt[i,j] * VVwin[ps+j, v],
    // Gt[i,j] = g[i][j-i] for 0<=j-i<23 else 0; j (band col) padded to K=40.
    for (int s = 0; s < 10; ++s) {
      const int jb = 4 * s + khi;
      v2f af;
      int c0 = jb + 0 - cN;
      int c1 = jb + 1 - cN;
      af.x = (c0 >= 0 && c0 < 23) ? sG[gbase + c0 * 24 + cN] : 0.f;
      af.y = (c1 >= 0 && c1 < 23) ? sG[gbase + c1 * 24 + cN] : 0.f;
      int r0 = ps + jb + 0; if (r0 > 85) r0 = 85;   // A==0 there; clamp row read
      int r1 = ps + jb + 1; if (r1 > 85) r1 = 85;
      #pragma unroll
      for (int vt = 0; vt < 4; ++vt) {
        v2f bf;
        bf.x = sVV[r0 * 80 + vt * 16 + cN];
        bf.y = sVV[r1 * 80 + vt * 16 + cN];
        acc[vt] = __builtin_amdgcn_wmma_f32_16x16x4_f32(
            false, af, false, bf, (short)0, acc[vt], false, false);
      }
    }

    // Stage to LDS for coalesced store: channel = h*64 + v
    #pragma unroll
    for (int vt = 0; vt < 4; ++vt) {
      #pragma unroll
      for (int r = 0; r < 8; ++r) {
        sOut[(h * 64 + vt * 16 + cN) * 68 + ps + r + hi8] = acc[vt][r];
      }
    }
  }
  __syncthreads();

  // Coalesced float4 store along n
  float* outp = out + (size_t)b * 256 * N_POS + n0;
  for (int idx = tid; idx < 256 * 16; idx += 256) {
    int ch = idx >> 4, p4 = (idx & 15) * 4;
    float4 val;
    val.x = sOut[ch * 68 + p4 + 0];
    val.y = sOut[ch * 68 + p4 + 1];
    val.z = sOut[ch * 68 + p4 + 2];
    val.w = sOut[ch * 68 + p4 + 3];
    *reinterpret_cast<float4*>(outp + (size_t)ch * N_POS + p4) = val;
  }
}

// ---------------------------------------------------------------------------
extern "C" void kernel_launch(void* const* d_in, const int* in_sizes, int n_in,
                              void* d_out, int out_size, void* d_ws, size_t ws_size,
                              hipStream_t stream) {
  (void)in_sizes; (void)n_in; (void)out_size; (void)ws_size;
  const float* x      = (const float*)d_in[0];
  const float* Wq     = (const float*)d_in[1];
  const float* q_g    = (const float*)d_in[2];
  const float* q_b    = (const float*)d_in[3];
  const float* q_m    = (const float*)d_in[4];
  const float* q_v    = (const float*)d_in[5];
  const float* Wk     = (const float*)d_in[6];
  const float* Wv     = (const float*)d_in[7];
  const float* v_g    = (const float*)d_in[8];
  const float* v_b    = (const float*)d_in[9];
  const float* v_m    = (const float*)d_in[10];
  const float* v_v    = (const float*)d_in[11];
  const float* conv_w = (const float*)d_in[12];
  const float* conv_b = (const float*)d_in[13];
  float* out = (float*)d_out;

  float* qbuf  = (float*)d_ws;                        // 16*64*4096
  float* vvbuf = qbuf  + (size_t)16 * 64 * N_POS;     // 16*64*4096
  float* kbuf  = vvbuf + (size_t)16 * 64 * N_POS;     // 16*16*4096
  float* lamc  = kbuf  + (size_t)16 * 16 * N_POS;     // 16*16*64

  proj_kernel<<<dim3(32, 16), 256, 0, stream>>>(
      x, Wq, Wk, Wv, q_g, q_b, q_m, q_v, v_g, v_b, v_m, v_v, qbuf, kbuf, vvbuf);
  softmax_kernel<<<dim3(256), 256, 0, stream>>>(kbuf);
  lamc_kernel<<<dim3(16), 256, 0, stream>>>(kbuf, vvbuf, conv_b, lamc);
  out_kernel<<<dim3(64, 16), 256, 0, stream>>>(qbuf, vvbuf, lamc, conv_w, out);
}